// SelfConvAtt_29085518529006
// MI455X (gfx1250) — compile-verified
//
#include <hip/hip_runtime.h>
#include <hip/hip_bf16.h>

typedef __attribute__((ext_vector_type(2))) float v2f;
typedef __attribute__((ext_vector_type(4))) float v4f;
typedef __attribute__((ext_vector_type(8))) float v8f;

#define BSZ  16
#define CDIM 384
#define HD   28
#define WD   28
#define SSZ  784          // 28*28
#define NPOS (BSZ * SSZ)  // 12544
#define NH   6
#define HDIM 64

// ---- async global->LDS path (guarded; falls back to b128 load/store if absent) ----
#if defined(__has_builtin)
#if __has_builtin(__builtin_amdgcn_global_load_async_to_lds_b128) && \
    __has_builtin(__builtin_amdgcn_s_wait_asynccnt)
#define USE_ASYNC_LDS 1
#endif
#endif
#ifndef USE_ASYNC_LDS
#define USE_ASYNC_LDS 0
#endif

typedef int v4i_t __attribute__((vector_size(16)));

__device__ __forceinline__ void copy16_g2lds(const float* g, float* l) {
#if USE_ASYNC_LDS
    __builtin_amdgcn_global_load_async_to_lds_b128(
        (__attribute__((address_space(1))) v4i_t*)(g),
        (__attribute__((address_space(3))) v4i_t*)(l), 0, 0);
#else
    *(v4f*)l = *(const v4f*)g;
#endif
}

__device__ __forceinline__ void wait_async_copies() {
#if USE_ASYNC_LDS
    __builtin_amdgcn_s_wait_asynccnt(0);
#endif
}

// -------- depthwise conv value at (b, h, w, c) for a given conv's channel weights --------
__device__ __forceinline__ float dw_val(const float* __restrict__ x1,
                                        const float* __restrict__ x2,
                                        int branch, int b, int h, int w, int c,
                                        const float* __restrict__ wts) {
    float acc = 0.f;
    #pragma unroll
    for (int dy = -1; dy <= 1; dy++) {
        int hh = h + dy;
        if (hh < 0 || hh >= HD) continue;
        #pragma unroll
        for (int dx = -1; dx <= 1; dx++) {
            int ww = w + dx;
            if (ww < 0 || ww >= WD) continue;
            size_t off = ((size_t)b * SSZ + (size_t)hh * WD + ww) * CDIM + c;
            float v;
            if (branch == 0)      v = x1[off];
            else if (branch == 1) v = x2[off];
            else                  v = x1[off] + x2[off];
            acc += v * wts[(dy + 1) * 3 + (dx + 1)];
        }
    }
    return acc;
}

// -------- BN partial sums: block = one batch, thread = one channel (coalesced) --------
__global__ __launch_bounds__(384)
void stats_part_kernel(const float* __restrict__ x1, const float* __restrict__ x2,
                       const float* __restrict__ dw_w, float* __restrict__ partial,
                       int conv) {
    int c = threadIdx.x;   // 0..383
    int b = blockIdx.x;    // 0..15
    int branch = conv / 3;
    float wts[9];
    const float* dwp = dw_w + ((size_t)conv * CDIM + c) * 9;
    #pragma unroll
    for (int i = 0; i < 9; i++) wts[i] = dwp[i];
    float sum = 0.f, sq = 0.f;
    for (int s = 0; s < SSZ; s++) {
        float y = dw_val(x1, x2, branch, b, s / WD, s % WD, c, wts);
        sum += y;
        sq += y * y;
    }
    partial[(size_t)(b * 2 + 0) * CDIM + c] = sum;
    partial[(size_t)(b * 2 + 1) * CDIM + c] = sq;
}

__global__ __launch_bounds__(384)
void stats_final_kernel(const float* __restrict__ partial, const float* __restrict__ gamma,
                        const float* __restrict__ beta, float* __restrict__ stats, int conv) {
    int c = threadIdx.x;
    float sum = 0.f, sq = 0.f;
    #pragma unroll
    for (int b = 0; b < BSZ; b++) {
        sum += partial[(size_t)(b * 2 + 0) * CDIM + c];
        sq  += partial[(size_t)(b * 2 + 1) * CDIM + c];
    }
    float mu  = sum / (float)NPOS;
    float var = sq / (float)NPOS - mu * mu;
    float sc  = gamma[conv * CDIM + c] * rsqrtf(var + 1e-5f);
    stats[((size_t)conv * CDIM + c) * 2 + 0] = sc;
    stats[((size_t)conv * CDIM + c) * 2 + 1] = beta[conv * CDIM + c] - mu * sc;
}

// -------- normalized depthwise output, written token-major (b, s, c) --------
__global__ __launch_bounds__(256)
void dwbn_kernel(const float* __restrict__ x1, const float* __restrict__ x2,
                 const float* __restrict__ dw_w, const float* __restrict__ stats,
                 float* __restrict__ Y, int conv) {
    int idx = blockIdx.x * blockDim.x + threadIdx.x;
    if (idx >= NPOS * CDIM) return;
    int c = idx % CDIM;
    int s = (idx / CDIM) % SSZ;
    int b = idx / (CDIM * SSZ);
    int branch = conv / 3;
    const float* dwp = dw_w + ((size_t)conv * CDIM + c) * 9;
    float wts[9];
    #pragma unroll
    for (int i = 0; i < 9; i++) wts[i] = dwp[i];
    float y  = dw_val(x1, x2, branch, b, s / WD, s % WD, c, wts);
    float sc = stats[((size_t)conv * CDIM + c) * 2 + 0];
    float bi = stats[((size_t)conv * CDIM + c) * 2 + 1];
    Y[idx] = y * sc + bi;
}

// -------- LDS-staged GEMM: D[m][n] = sum_k A[m][k] * W[n][k] (+ bias[n]) --------
// Block tile 64(M) x 32(N), 8 waves = 4x2 subtiles of 16x16; K chunks of 32 staged
// in LDS via async global->LDS (row stride 34 floats: conflict-free ds_load_b64).
#define KC  32
#define LDS_STRIDE 34

__global__ __launch_bounds__(256)
void gemm_wmma(const float* __restrict__ A, const float* __restrict__ W,
               const float* __restrict__ bias, float* __restrict__ D, int hasBias) {
    __shared__ __align__(16) float As[64 * LDS_STRIDE];
    __shared__ __align__(16) float Ws[32 * LDS_STRIDE];
    int tid  = threadIdx.x;
    int wave = tid >> 5, lane = tid & 31;
    int wm = wave >> 1, wn = wave & 1;          // 4 x 2 subtile grid
    int m0 = blockIdx.x * 64;
    int n0 = blockIdx.y * 32;
    int m16 = lane & 15;
    int khalf = (lane >> 4) * 2;                // K offset pair within a 4-step

    v8f c = {};
    for (int kc = 0; kc < CDIM; kc += KC) {
        __syncthreads();   // previous chunk's fragment reads done before overwrite
        // stage A: 64 rows x 32 floats = 512 x 16B items, 2 per thread
        #pragma unroll
        for (int it = 0; it < 2; it++) {
            int i = tid + it * 256;
            int r = i >> 3, g = i & 7;
            copy16_g2lds(A + (size_t)(m0 + r) * CDIM + kc + g * 4,
                         &As[r * LDS_STRIDE + g * 4]);
        }
        // stage W: 32 rows x 32 floats = 256 x 16B items, 1 per thread
        {
            int r = tid >> 3, g = tid & 7;
            copy16_g2lds(W + (size_t)(n0 + r) * CDIM + kc + g * 4,
                         &Ws[r * LDS_STRIDE + g * 4]);
        }
        wait_async_copies();
        __syncthreads();
        #pragma unroll
        for (int kk = 0; kk < KC; kk += 4) {
            v2f a = *(const v2f*)&As[(wm * 16 + m16) * LDS_STRIDE + kk + khalf];
            v2f b = *(const v2f*)&Ws[(wn * 16 + m16) * LDS_STRIDE + kk + khalf];
            c = __builtin_amdgcn_wmma_f32_16x16x4_f32(false, a, false, b, (short)0, c,
                                                      false, false);
        }
    }
    float bv = hasBias ? bias[n0 + wn * 16 + m16] : 0.f;
    int mbase = m0 + wm * 16 + ((lane >> 4) << 3);  // row r -> m = r + 8*(lane/16)
    int ncol  = n0 + wn * 16 + m16;
    #pragma unroll
    for (int r = 0; r < 8; r++)
        D[(size_t)(mbase + r) * CDIM + ncol] = c[r] + bv;
}

// -------- Flash attention: one wave per 16-query tile per (b, h) --------
__global__ __launch_bounds__(32)
void attn_kernel(const float* __restrict__ Q, const float* __restrict__ K,
                 const float* __restrict__ V, float* __restrict__ O) {
    __shared__ __align__(16) float pbuf[16 * 16];
    int lane = threadIdx.x;
    int qt = blockIdx.x;             // 0..48
    int bh = blockIdx.y;             // 0..95
    int b = bh / NH, h = bh % NH;
    const float* Qb = Q + (size_t)b * SSZ * CDIM + h * HDIM;
    const float* Kb = K + (size_t)b * SSZ * CDIM + h * HDIM;
    const float* Vb = V + (size_t)b * SSZ * CDIM + h * HDIM;

    int m16 = lane & 15;
    int khalf = (lane >> 4) * 2;
    const float scale = 0.05103103630798287f;   // 384^-0.5 (reference scales by C)

    v2f qa[16];
    const float* qrow = Qb + (size_t)(qt * 16 + m16) * CDIM + khalf;
    #pragma unroll
    for (int kk = 0; kk < 16; kk++) qa[kk] = *(const v2f*)(qrow + kk * 4);

    float mi[8], li[8];
    v8f acc0 = {}, acc1 = {}, acc2 = {}, acc3 = {};
    #pragma unroll
    for (int r = 0; r < 8; r++) { mi[r] = -1e30f; li[r] = 0.f; }

    for (int kt = 0; kt < 49; kt++) {
        v8f sc = {};
        const float* krow = Kb + (size_t)(kt * 16 + m16) * CDIM + khalf;
        #pragma unroll
        for (int kk = 0; kk < 16; kk++) {
            v2f bf = *(const v2f*)(krow + kk * 4);
            sc = __builtin_amdgcn_wmma_f32_16x16x4_f32(false, qa[kk], false, bf,
                                                       (short)0, sc, false, false);
        }
        float alpha[8];
        #pragma unroll
        for (int r = 0; r < 8; r++) {
            float v = sc[r] * scale;
            float mx = v;
            #pragma unroll
            for (int off = 1; off < 16; off <<= 1)
                mx = fmaxf(mx, __shfl_xor(mx, off, 32));
            float mnew = fmaxf(mi[r], mx);
            alpha[r] = expf(mi[r] - mnew);
            float p = expf(v - mnew);
            float rs = p;
            #pragma unroll
            for (int off = 1; off < 16; off <<= 1)
                rs += __shfl_xor(rs, off, 32);
            li[r] = li[r] * alpha[r] + rs;
            mi[r] = mnew;
            sc[r] = p;
        }
        #pragma unroll
        for (int r = 0; r < 8; r++) {
            acc0[r] *= alpha[r]; acc1[r] *= alpha[r];
            acc2[r] *= alpha[r]; acc3[r] *= alpha[r];
        }
        __syncthreads();
        #pragma unroll
        for (int r = 0; r < 8; r++)
            pbuf[(r + ((lane >> 4) << 3)) * 16 + m16] = sc[r];
        __syncthreads();
        #pragma unroll
        for (int kk2 = 0; kk2 < 4; kk2++) {
            v2f pa = *(const v2f*)&pbuf[m16 * 16 + kk2 * 4 + khalf];
            const float* vp = Vb + (size_t)(kt * 16 + kk2 * 4 + khalf) * CDIM + m16;
            v2f b0, b1, b2, b3;
            b0.x = vp[0];  b0.y = vp[CDIM + 0];
            b1.x = vp[16]; b1.y = vp[CDIM + 16];
            b2.x = vp[32]; b2.y = vp[CDIM + 32];
            b3.x = vp[48]; b3.y = vp[CDIM + 48];
            acc0 = __builtin_amdgcn_wmma_f32_16x16x4_f32(false, pa, false, b0, (short)0, acc0, false, false);
            acc1 = __builtin_amdgcn_wmma_f32_16x16x4_f32(false, pa, false, b1, (short)0, acc1, false, false);
            acc2 = __builtin_amdgcn_wmma_f32_16x16x4_f32(false, pa, false, b2, (short)0, acc2, false, false);
            acc3 = __builtin_amdgcn_wmma_f32_16x16x4_f32(false, pa, false, b3, (short)0, acc3, false, false);
        }
    }
    float* Ob = O + (size_t)b * SSZ * CDIM + (size_t)h * SSZ * HDIM;
    #pragma unroll
    for (int r = 0; r < 8; r++) {
        float inv = 1.f / li[r];
        int t = qt * 16 + r + ((lane >> 4) << 3);
        float* op = Ob + (size_t)t * HDIM + m16;
        op[0]  = acc0[r] * inv;
        op[16] = acc1[r] * inv;
        op[32] = acc2[r] * inv;
        op[48] = acc3[r] * inv;
    }
}

extern "C" void kernel_launch(void* const* d_in, const int* in_sizes, int n_in,
                              void* d_out, int out_size, void* d_ws, size_t ws_size,
                              hipStream_t stream) {
    const float* x1    = (const float*)d_in[0];
    const float* x2    = (const float*)d_in[3];
    const float* dw_w  = (const float*)d_in[6];
    const float* gamma = (const float*)d_in[7];
    const float* beta  = (const float*)d_in[8];
    const float* pw_w  = (const float*)d_in[9];
    const float* pw_b  = (const float*)d_in[10];
    const float* lin_w = (const float*)d_in[11];
    float* out = (float*)d_out;

    const size_t plane = (size_t)NPOS * CDIM;  // 4,816,896 floats
    float* ws      = (float*)d_ws;
    float* stats   = ws;                       // 9*384*2 floats
    float* partial = ws + 8192;                // 16*2*384 floats
    float* Y       = ws + 32768;
    float* P       = Y + plane;
    float* Lq      = P + plane;
    float* Lk      = Lq + plane;
    float* Lv      = Lk + plane;
    float* L[3]    = {Lq, Lk, Lv};

    for (int br = 0; br < 3; br++) {
        for (int q = 0; q < 3; q++) {
            int conv = br * 3 + q;
            stats_part_kernel<<<BSZ, CDIM, 0, stream>>>(x1, x2, dw_w, partial, conv);
            stats_final_kernel<<<1, CDIM, 0, stream>>>(partial, gamma, beta, stats, conv);
            dwbn_kernel<<<(NPOS * CDIM) / 256, 256, 0, stream>>>(x1, x2, dw_w, stats, Y, conv);
            gemm_wmma<<<dim3(NPOS / 64, CDIM / 32), 256, 0, stream>>>(
                Y, pw_w + (size_t)conv * CDIM * CDIM, pw_b + (size_t)conv * CDIM, P, 1);
            int li = ((br == 2) ? 3 : 3 * br) + q;
            gemm_wmma<<<dim3(NPOS / 64, CDIM / 32), 256, 0, stream>>>(
                P, lin_w + (size_t)li * CDIM * CDIM, nullptr, L[q], 0);
        }
        attn_kernel<<<dim3(SSZ / 16, BSZ * NH), 32, 0, stream>>>(
            Lq, Lk, Lv, out + (size_t)br * plane);
    }
}